// TranslationNet_90658169684623
// MI455X (gfx1250) — compile-verified
//
#include <hip/hip_runtime.h>

// out[i,j] = x[i,j] + params[j]   (N=16384 rows, D=4096 cols, f32)
//
// Streaming workload: ~537 MB per call -> ~23 us floor at 23.3 TB/s HBM.
// Broadcast-add computed as a rank-1 WMMA update (A = ones column, B row0 =
// params chunk, C = x tile) via V_WMMA_F32_16X16X4_F32 — bit-exact vs. the
// reference f32 add. x/out use non-temporal hints (512 MiB > 192 MB L2).
//
// D is a template constant so the 8 row strides (r * 16 KiB <= 112 KiB)
// fold into the 24-bit IOFFSET of each global load/store: one base address,
// one s_clause of 8 NT loads, wmma, one s_clause of 8 NT stores.

typedef __attribute__((ext_vector_type(2))) float v2f;
typedef __attribute__((ext_vector_type(8))) float v8f;

template <int DC>
__global__ __launch_bounds__(256) void TranslationNet_bcast_add_wmma(
    const float* __restrict__ x, const float* __restrict__ p,
    float* __restrict__ out, int d_rt) {
  const int D      = (DC > 0) ? DC : d_rt;
  const int lane   = threadIdx.x & 31;
  const int wave   = threadIdx.x >> 5;          // 8 wave32s per block
  const int col0   = blockIdx.x * 128 + wave * 16;
  const int row0   = blockIdx.y * 16;
  const int lane_n = lane & 15;                  // N index within tile
  const int mhalf  = (lane >> 4) << 3;           // lanes 16-31 -> rows M+8

  // B (4x16 f32, 2 VGPRs): row K=0 = params chunk on lanes 0-15, rest 0.
  float pv = p[col0 + lane_n];                   // coalesced 64B load
  v2f b; b.x = (lane < 16) ? pv : 0.0f; b.y = 0.0f;

  // A (16x4 f32, 2 VGPRs): column K=0 all ones, K=1..3 zero.
  v2f a; a.x = (lane < 16) ? 1.0f : 0.0f; a.y = 0.0f;

  // C/D 16x16 f32 layout: VGPR r = row M=r (lanes 0-15) / M=r+8 (lanes
  // 16-31), N = lane&15. Each access = 2 contiguous 64B segments per wave.
  const size_t base = (size_t)(row0 + mhalf) * (size_t)D + (size_t)(col0 + lane_n);
  const float* xb = x + base;
  float*       ob = out + base;

  v8f c;
#pragma unroll
  for (int r = 0; r < 8; ++r)
    c[r] = __builtin_nontemporal_load(xb + r * D);   // IOFFSET = r*D*4 bytes

  // D = A*B + C = x_tile + broadcast(params_chunk)
  v8f d = __builtin_amdgcn_wmma_f32_16x16x4_f32(
      /*neg_a=*/false, a, /*neg_b=*/false, b,
      /*c_mod=*/(short)0, c, /*reuse_a=*/false, /*reuse_b=*/false);

#pragma unroll
  for (int r = 0; r < 8; ++r)
    __builtin_nontemporal_store(d[r], ob + r * D);

  (void)d_rt;
}

extern "C" void kernel_launch(void* const* d_in, const int* in_sizes, int n_in,
                              void* d_out, int out_size, void* d_ws, size_t ws_size,
                              hipStream_t stream) {
  const float* x = (const float*)d_in[0];
  const float* p = (const float*)d_in[1];
  float* out     = (float*)d_out;

  const int D = in_sizes[1];             // 4096
  const int N = in_sizes[0] / D;         // 16384

  dim3 block(256, 1, 1);                 // 8 wave32s, one 16x16 tile each
  dim3 grid(D / 128, N / 16, 1);         // 32 x 1024 blocks

  if (D == 4096) {
    TranslationNet_bcast_add_wmma<4096><<<grid, block, 0, stream>>>(x, p, out, D);
  } else {
    TranslationNet_bcast_add_wmma<0><<<grid, block, 0, stream>>>(x, p, out, D);
  }
}